// DNC_56203942035984
// MI455X (gfx1250) — compile-verified
//
#include <hip/hip_runtime.h>
#include <math.h>

// ---------------- problem constants ----------------
#define BATCH 32
#define TLEN  64
#define INDIM 256
#define HDIM  512
#define G4H   2048      // 4*H
#define MSLOT 128
#define WWORD 64
#define RHEAD 4
#define IFACE 471
#define IFPAD 480
#define HRW   768       // H + R*W
#define CLIPV 20.0f
#define EPSV  1e-6f
#define ADEL  5e-6f

typedef __attribute__((ext_vector_type(2))) float v2f;
typedef __attribute__((ext_vector_type(8))) float v8f;

__device__ __forceinline__ float sigmoidf_(float x) { return 1.0f / (1.0f + expf(-x)); }
__device__ __forceinline__ float softplusf_(float x) {
    return (x > 0.0f) ? (x + log1pf(expf(-x))) : log1pf(expf(x));
}

// ---------------- fp32 WMMA K-loop: C += A(16xK) * W(KxN=16)^T -------------
// ap: pointer to A row (row0+li), already offset by 2*half
// wp: pointer to W row (col0+li), already offset by 2*half  (W is [N,K] row-major)
__device__ __forceinline__ void mma_f32_loop(const float* __restrict__ ap,
                                             const float* __restrict__ wp,
                                             int K, v8f& acc) {
#pragma unroll 4
    for (int k = 0; k < K; k += 4) {
        v2f a; a.x = ap[k]; a.y = ap[k + 1];
        v2f w; w.x = wp[k]; w.y = wp[k + 1];
        acc = __builtin_amdgcn_wmma_f32_16x16x4_f32(false, a, false, w,
                                                    (short)0, acc, false, false);
    }
}

// ---------------- LSTM gate GEMM: G[32,2048] = A1*W1^T + A2*W2^T + b1 + b2 --
__global__ __launch_bounds__(256) void dnc_gates_kernel(
    const float* __restrict__ A1, const float* __restrict__ W1,
    const float* __restrict__ A2, const float* __restrict__ W2,
    const float* __restrict__ b1, const float* __restrict__ b2,
    float* __restrict__ G) {
    const int lane = threadIdx.x & 31;
    const int wid  = blockIdx.x * 8 + (threadIdx.x >> 5);   // 256 waves
    const int nt = wid & 127, mt = wid >> 7;
    const int row0 = mt * 16, col0 = nt * 16;
    const int half = lane >> 4, li = lane & 15;

    v8f acc = {};
    mma_f32_loop(A1 + (row0 + li) * HDIM + 2 * half,
                 W1 + (col0 + li) * HDIM + 2 * half, HDIM, acc);
    mma_f32_loop(A2 + (row0 + li) * HDIM + 2 * half,
                 W2 + (col0 + li) * HDIM + 2 * half, HDIM, acc);

    const int col = col0 + li;
    const float bb = b1[col] + b2[col];
    float* gout = G + (row0 + 8 * half) * G4H + col;
#pragma unroll
    for (int j = 0; j < 8; ++j) gout[j * G4H] = acc[j] + bb;
}

// ---------------- interface GEMM: V[32,480] = A*W_if^T + b_if ---------------
__global__ __launch_bounds__(256) void dnc_iface_kernel(
    const float* __restrict__ A, const float* __restrict__ Wif,
    const float* __restrict__ bif, float* __restrict__ V) {
    const int wid = blockIdx.x * 8 + (threadIdx.x >> 5);
    if (wid >= 60) return;                                  // 2 x 30 tiles
    const int lane = threadIdx.x & 31;
    const int mt = wid / 30, nt = wid % 30;
    const int row0 = mt * 16, col0 = nt * 16;
    const int half = lane >> 4, li = lane & 15;
    const int col = col0 + li;
    const int wrow = (col < IFACE) ? col : (IFACE - 1);     // clamp OOB W rows

    v8f acc = {};
    mma_f32_loop(A + (row0 + li) * HDIM + 2 * half,
                 Wif + wrow * HDIM + 2 * half, HDIM, acc);

    if (col < IFACE) {
        const float bb = bif[col];
        float* vout = V + (row0 + 8 * half) * IFPAD + col;
#pragma unroll
        for (int j = 0; j < 8; ++j) vout[j * IFPAD] = acc[j] + bb;
    }
}

// ---------------- output projection: Y = outs[2048,768] * W_out^T + b ------
__global__ __launch_bounds__(256) void dnc_proj_kernel(
    const float* __restrict__ outs, const float* __restrict__ Wout,
    const float* __restrict__ bout, float* __restrict__ Y) {
    const int wid  = blockIdx.x * 8 + (threadIdx.x >> 5);   // 2048 waves
    const int lane = threadIdx.x & 31;
    const int mt = wid >> 4, nt = wid & 15;
    const int row0 = mt * 16, col0 = nt * 16;
    const int half = lane >> 4, li = lane & 15;

    v8f acc = {};
    mma_f32_loop(outs + (row0 + li) * HRW + 2 * half,
                 Wout + (col0 + li) * HRW + 2 * half, HRW, acc);

    const int col = col0 + li;
    const float bb = bout[col];
#pragma unroll
    for (int j = 0; j < 8; ++j) {
        const int row = row0 + 8 * half + j;                // row = t*32 + b
        const int t = row >> 5, b = row & 31;
        Y[(b * TLEN + t) * INDIM + col] = acc[j] + bb;      // [B,T,IN]
    }
}

// ---------------- build LSTM0 input (last_read is a closure var == 0) ------
__global__ __launch_bounds__(512) void dnc_build_inp_kernel(
    const float* __restrict__ x, float* __restrict__ inp, int t) {
    const int i = blockIdx.x * 512 + threadIdx.x;           // 32*512
    const int b = i >> 9, k = i & 511;
    inp[i] = (k < INDIM) ? x[(b * TLEN + t) * INDIM + k] : 0.0f;
}

// ---------------- LSTM cell nonlinearity -----------------------------------
__global__ __launch_bounds__(512) void dnc_cell_kernel(
    const float* __restrict__ G, float* __restrict__ h, float* __restrict__ c,
    float* __restrict__ outb, float* __restrict__ outs_t) {
    const int i = blockIdx.x * 512 + threadIdx.x;           // 32*512
    const int b = i >> 9, j = i & 511;
    const float* g = G + b * G4H;
    const float gi = g[j], gf = g[HDIM + j], gg = g[2 * HDIM + j], go = g[3 * HDIM + j];
    const float cn = sigmoidf_(gf) * c[i] + sigmoidf_(gi) * tanhf(gg);
    const float hn = sigmoidf_(go) * tanhf(cn);
    c[i] = cn; h[i] = hn;
    if (outb != nullptr) {
        const float ov = fminf(fmaxf(hn, -CLIPV), CLIPV);
        outb[i] = ov;
        outs_t[b * HRW + j] = ov;
    }
}

// ---------------- state init ------------------------------------------------
__global__ __launch_bounds__(256) void dnc_init_kernel(
    const float* __restrict__ hin, float* h0s, float* c0s, float* h1s, float* c1s,
    float* mem, float* lnk, float* prc, float* rwst, float* wwst, float* usg) {
    const int i = blockIdx.x * 256 + threadIdx.x;           // 524288 threads
    if (i < BATCH * HDIM) {
        const float a = hin[i], b = hin[BATCH * HDIM + i];
        h0s[i] = a; c0s[i] = a; h1s[i] = b; c1s[i] = b;
        rwst[i] = 0.0f;                                     // 32*4*128 == 16384
    }
    if (i < BATCH * MSLOT * WWORD) mem[i] = 0.0f;
    lnk[i] = 0.0f;                                          // 32*128*128
    if (i < BATCH * MSLOT) { prc[i] = 0.0f; wwst[i] = 0.0f; usg[i] = 0.0f; }
}

// ---------------- block reductions (128 threads) ---------------------------
__device__ __forceinline__ float blk_max128(float v, float* s) {
    const int tid = threadIdx.x;
    s[tid] = v; __syncthreads();
    for (int off = 64; off > 0; off >>= 1) {
        if (tid < off) s[tid] = fmaxf(s[tid], s[tid + off]);
        __syncthreads();
    }
    const float r = s[0]; __syncthreads();
    return r;
}
__device__ __forceinline__ float blk_sum128(float v, float* s) {
    const int tid = threadIdx.x;
    s[tid] = v; __syncthreads();
    for (int off = 64; off > 0; off >>= 1) {
        if (tid < off) s[tid] = s[tid] + s[tid + off];
        __syncthreads();
    }
    const float r = s[0]; __syncthreads();
    return r;
}

// ---------------- DNC memory controller: one block (128 thr) per batch -----
__global__ __launch_bounds__(128) void dnc_memstep_kernel(
    const float* __restrict__ vbuf, float* __restrict__ mem,
    float* __restrict__ link, float* __restrict__ prc,
    float* __restrict__ rwst, float* __restrict__ wwst,
    float* __restrict__ usg, float* __restrict__ outs_t) {
    const int b = blockIdx.x, tid = threadIdx.x;

    __shared__ float vt[IFPAD];
    __shared__ float s_modes[RHEAD * 3];
    __shared__ float s_rwp[RHEAD][MSLOT];
    __shared__ float s_cw[RHEAD][MSLOT];
    __shared__ float s_rwn[RHEAD][MSLOT];
    __shared__ float s_wcw[MSLOT];
    __shared__ float s_alloc[MSLOT];
    __shared__ float s_ww[MSLOT];
    __shared__ float skey[MSLOT];
    __shared__ int   sidx[MSLOT];
    __shared__ float s_pre[MSLOT];
    __shared__ float s_red[MSLOT];

    const float* vb = vbuf + b * IFPAD;
    float* memb = mem  + b * MSLOT * WWORD;
    float* lnk  = link + b * MSLOT * MSLOT;

    // ---- stage + transform interface vector ----
    for (int i = tid; i < IFACE; i += 128) {
        const float x = vb[i]; float y;
        if      (i < 256) y = tanhf(x);          // read keys
        else if (i < 260) y = softplusf_(x);     // read strengths
        else if (i < 324) y = tanhf(x);          // write key
        else if (i < 325) y = softplusf_(x);     // write strength
        else if (i < 389) y = sigmoidf_(x);      // erase vec
        else if (i < 453) y = tanhf(x);          // write vec
        else if (i < 459) y = sigmoidf_(x);      // fg[4], ag, wg
        else              y = x;                 // raw mode logits
        vt[i] = y;
    }
    __syncthreads();
    if (tid < RHEAD) {                           // mode softmax (3-way)
        const float a = vt[459 + 3 * tid], bb = vt[460 + 3 * tid], cc = vt[461 + 3 * tid];
        const float mx = fmaxf(a, fmaxf(bb, cc));
        const float ea = expf(a - mx), eb = expf(bb - mx), ec = expf(cc - mx);
        const float s = ea + eb + ec;
        s_modes[3 * tid + 0] = ea / s;
        s_modes[3 * tid + 1] = eb / s;
        s_modes[3 * tid + 2] = ec / s;
    }
    for (int i = tid; i < RHEAD * MSLOT; i += 128)
        s_rwp[i >> 7][i & 127] = rwst[b * RHEAD * MSLOT + i];
    s_pre[tid] = prc[b * MSLOT + tid];
    __syncthreads();

    // ---- usage update ----
    float us  = usg[b * MSLOT + tid];
    const float wwp = wwst[b * MSLOT + tid];
    us = us + (1.0f - us) * wwp;
    float fpr = 1.0f;
#pragma unroll
    for (int r = 0; r < RHEAD; ++r) fpr *= (1.0f - vt[453 + r] * s_rwp[r][tid]);
    us *= fpr;
    usg[b * MSLOT + tid] = us;

    // ---- write content weighting (old memory) ----
    float dot = 0.0f, mm = 0.0f, kk = 0.0f;
    for (int w = 0; w < WWORD; ++w) {
        const float mv = memb[tid * WWORD + w];
        const float kv = vt[260 + w];
        dot += mv * kv; mm += mv * mv; kk += kv * kv;
    }
    float score = dot / ((sqrtf(mm) + EPSV) * (sqrtf(kk) + EPSV)) * vt[324];
    {
        const float mx = blk_max128(score, s_red);
        const float e  = expf(score - mx);
        const float ss = blk_sum128(e, s_red);
        s_wcw[tid] = e / ss;
    }

    // ---- allocation weighting: stable argsort of u ----
    const float u = ADEL + (1.0f - ADEL) * us;
    skey[tid] = u; sidx[tid] = tid; __syncthreads();
    for (int k2 = 2; k2 <= MSLOT; k2 <<= 1) {
        for (int j2 = k2 >> 1; j2 > 0; j2 >>= 1) {
            const int ixj = tid ^ j2;
            if (ixj > tid) {
                const float ka = skey[tid], kb = skey[ixj];
                const int   ia = sidx[tid], ib = sidx[ixj];
                const bool agtb = (ka > kb) || (ka == kb && ia > ib);
                const bool up = ((tid & k2) == 0);
                if (up ? agtb : !agtb) {
                    skey[tid] = kb; skey[ixj] = ka;
                    sidx[tid] = ib; sidx[ixj] = ia;
                }
            }
            __syncthreads();
        }
    }
    {   // exclusive prefix product of sorted usage
        float p = 1.0f;
        for (int i2 = 0; i2 < tid; ++i2) p *= skey[i2];
        const float sa = (1.0f - skey[tid]) * p;
        s_alloc[sidx[tid]] = sa;
    }
    __syncthreads();

    // ---- write weighting ----
    const float ag = vt[457], wg = vt[458];
    const float wwm = wg * (ag * s_alloc[tid] + (1.0f - ag) * s_wcw[tid]);
    s_ww[tid] = wwm;
    const float wwsum = blk_sum128(wwm, s_red);   // includes syncthreads

    // ---- memory erase + write ----
    for (int w = 0; w < WWORD; ++w) {
        float mv = memb[tid * WWORD + w];
        mv = mv * (1.0f - wwm * vt[325 + w]) + wwm * vt[389 + w];
        memb[tid * WWORD + w] = mv;
    }

    // ---- link matrix update (old precedence) ----
    for (int e2 = tid; e2 < MSLOT * MSLOT; e2 += 128) {
        const int i2 = e2 >> 7, j2 = e2 & 127;
        float Lv = lnk[e2];
        Lv = (1.0f - s_ww[i2] - s_ww[j2]) * Lv + s_ww[i2] * s_pre[j2];
        if (i2 == j2) Lv = 0.0f;
        lnk[e2] = Lv;
    }
    prc[b * MSLOT + tid] = (1.0f - wwsum) * s_pre[tid] + s_ww[tid];
    __syncthreads();   // link + memory globally visible within block

    // ---- read content weighting (new memory) ----
    float dr[RHEAD] = {0.0f, 0.0f, 0.0f, 0.0f};
    float mm2 = 0.0f;
    for (int w = 0; w < WWORD; ++w) {
        const float mv = memb[tid * WWORD + w];
        mm2 += mv * mv;
#pragma unroll
        for (int r = 0; r < RHEAD; ++r) dr[r] += mv * vt[r * WWORD + w];
    }
#pragma unroll
    for (int r = 0; r < RHEAD; ++r) {
        float kn = 0.0f;
        for (int w = 0; w < WWORD; ++w) { const float kv = vt[r * WWORD + w]; kn += kv * kv; }
        const float sc = dr[r] / ((sqrtf(mm2) + EPSV) * (sqrtf(kn) + EPSV)) * vt[256 + r];
        const float mx = blk_max128(sc, s_red);
        const float e  = expf(sc - mx);
        const float ss = blk_sum128(e, s_red);
        s_cw[r][tid] = e / ss;
    }

    // ---- forward / backward weights through link ----
    float fw[RHEAD] = {0.0f, 0.0f, 0.0f, 0.0f};
    float bw[RHEAD] = {0.0f, 0.0f, 0.0f, 0.0f};
    for (int q = 0; q < MSLOT; ++q) {
        const float Lmq = lnk[tid * MSLOT + q];
        const float Lqm = lnk[q * MSLOT + tid];
#pragma unroll
        for (int r = 0; r < RHEAD; ++r) {
            fw[r] += Lmq * s_rwp[r][q];
            bw[r] += s_rwp[r][q] * Lqm;
        }
    }
#pragma unroll
    for (int r = 0; r < RHEAD; ++r) {
        const float rv = s_modes[3 * r + 0] * bw[r]
                       + s_modes[3 * r + 1] * fw[r]
                       + s_modes[3 * r + 2] * s_cw[r][tid];
        s_rwn[r][tid] = rv;
        rwst[b * RHEAD * MSLOT + r * MSLOT + tid] = rv;
    }
    wwst[b * MSLOT + tid] = wwm;
    __syncthreads();

    // ---- read vectors -> outs[t, b, 512:768] ----
    for (int o = tid; o < RHEAD * WWORD; o += 128) {
        const int r = o >> 6, w = o & 63;
        float acc = 0.0f;
        for (int m2 = 0; m2 < MSLOT; ++m2) acc += s_rwn[r][m2] * memb[m2 * WWORD + w];
        outs_t[b * HRW + HDIM + o] = acc;
    }
}

// ---------------- host-side orchestration ----------------------------------
extern "C" void kernel_launch(void* const* d_in, const int* in_sizes, int n_in,
                              void* d_out, int out_size, void* d_ws, size_t ws_size,
                              hipStream_t stream) {
    const float* x     = (const float*)d_in[0];
    const float* h0in  = (const float*)d_in[1];
    const float* W_ih0 = (const float*)d_in[2];
    const float* W_hh0 = (const float*)d_in[3];
    const float* b_ih0 = (const float*)d_in[4];
    const float* b_hh0 = (const float*)d_in[5];
    const float* W_ih1 = (const float*)d_in[6];
    const float* W_hh1 = (const float*)d_in[7];
    const float* b_ih1 = (const float*)d_in[8];
    const float* b_hh1 = (const float*)d_in[9];
    const float* W_if  = (const float*)d_in[10];
    const float* b_if  = (const float*)d_in[11];
    const float* W_out = (const float*)d_in[12];
    const float* b_out = (const float*)d_in[13];
    float* Y = (float*)d_out;

    float* ws = (float*)d_ws;
    size_t off = 0;
    float* inp  = ws + off; off += (size_t)BATCH * HDIM;          // 16384
    float* g0   = ws + off; off += (size_t)BATCH * G4H;           // 65536
    float* g1   = ws + off; off += (size_t)BATCH * G4H;           // 65536
    float* h0s  = ws + off; off += (size_t)BATCH * HDIM;
    float* c0s  = ws + off; off += (size_t)BATCH * HDIM;
    float* h1s  = ws + off; off += (size_t)BATCH * HDIM;
    float* c1s  = ws + off; off += (size_t)BATCH * HDIM;
    float* outb = ws + off; off += (size_t)BATCH * HDIM;
    float* vbuf = ws + off; off += (size_t)BATCH * IFPAD;         // 15360
    float* memS = ws + off; off += (size_t)BATCH * MSLOT * WWORD; // 262144
    float* lnkS = ws + off; off += (size_t)BATCH * MSLOT * MSLOT; // 524288
    float* prcS = ws + off; off += (size_t)BATCH * MSLOT;
    float* rwS  = ws + off; off += (size_t)BATCH * RHEAD * MSLOT;
    float* wwS  = ws + off; off += (size_t)BATCH * MSLOT;
    float* usS  = ws + off; off += (size_t)BATCH * MSLOT;
    float* outs = ws + off; off += (size_t)TLEN * BATCH * HRW;    // 1572864

    // state init (must run every call; harness does not re-poison between replays)
    dnc_init_kernel<<<2048, 256, 0, stream>>>(h0in, h0s, c0s, h1s, c1s,
                                              memS, lnkS, prcS, rwS, wwS, usS);

    for (int t = 0; t < TLEN; ++t) {
        float* outs_t = outs + (size_t)t * BATCH * HRW;
        dnc_build_inp_kernel<<<32, 512, 0, stream>>>(x, inp, t);
        dnc_gates_kernel<<<32, 256, 0, stream>>>(inp, W_ih0, h0s, W_hh0,
                                                 b_ih0, b_hh0, g0);
        dnc_cell_kernel<<<32, 512, 0, stream>>>(g0, h0s, c0s, nullptr, nullptr);
        dnc_gates_kernel<<<32, 256, 0, stream>>>(h0s, W_ih1, h1s, W_hh1,
                                                 b_ih1, b_hh1, g1);
        dnc_cell_kernel<<<32, 512, 0, stream>>>(g1, h1s, c1s, outb, outs_t);
        dnc_iface_kernel<<<8, 256, 0, stream>>>(outb, W_if, b_if, vbuf);
        dnc_memstep_kernel<<<32, 128, 0, stream>>>(vbuf, memS, lnkS, prcS,
                                                   rwS, wwS, usS, outs_t);
    }

    // final projection over all timesteps: [2048,768] x [256,768]^T
    dnc_proj_kernel<<<256, 256, 0, stream>>>(outs, W_out, b_out, Y);
}